// MyModel_83580063580175
// MI455X (gfx1250) — compile-verified
//
#include <hip/hip_runtime.h>
#include <math.h>
#include <limits.h>

// ---------------------------------------------------------------------------
// Problem constants (from reference)
// ---------------------------------------------------------------------------
#define HDIM 256
#define WDIM 256
#define NMAPS 512            // 32 * 16
#define RAD   12             // int(TRUNCATE*SIGMA + 0.5) = 12
#define KLEN  25             // 2*RAD+1
#define THRESH 0.01f
#define OFFSET_C 0.4395f
#define SCORE_THRESH 0.8f

typedef __attribute__((ext_vector_type(2))) float v2f;
typedef __attribute__((ext_vector_type(8))) float v8f;
typedef __attribute__((__vector_size__(4 * sizeof(int)))) int i32x4;
typedef __attribute__((address_space(1))) i32x4 glob_i32x4;
typedef __attribute__((address_space(3))) i32x4 lds_i32x4;

// ---------------------------------------------------------------------------
// Kernel 1: build the 256x256 Gaussian blur matrix A (and A^T) with symmetric
// padding folded into boundary columns.  out[w] = sum_src in[src] * A[src][w]
// ---------------------------------------------------------------------------
__global__ void build_blur_matrix(float* __restrict__ A, float* __restrict__ At) {
    const int i = blockIdx.x;    // source index (row of A)
    const int j = threadIdx.x;   // output index (col of A)
    float ksum = 0.0f;
    for (int t = 0; t < KLEN; ++t) {
        float off = (float)(t - RAD);
        ksum += __expf(-(off * off) * (1.0f / 18.0f));   // sigma=3 -> 2*s^2=18
    }
    float acc = 0.0f;
    for (int t = 0; t < KLEN; ++t) {
        int q = j + t - RAD;
        if (q < 0)      q = -1 - q;         // symmetric reflect (includes edge)
        if (q > WDIM-1) q = 2*WDIM - 1 - q;
        if (q == i) {
            float off = (float)(t - RAD);
            acc += __expf(-(off * off) * (1.0f / 18.0f));
        }
    }
    const float val = acc / ksum;
    A [i * WDIM + j] = val;
    At[j * WDIM + i] = val;
}

// ---------------------------------------------------------------------------
// Kernel 2: per-map blur as two GEMMs on the WMMA pipe.
//   T1 = m * A      (horizontal blur)   -> LDS stripe
//   G  = A^T * T1   (vertical blur)     -> global workspace
// One workgroup (512 threads = 16 waves) per map.  Map staged once into LDS
// via async-to-LDS; A column stripes staged per-jb into LDS in a K-paired
// layout so each B-fragment is one contiguous v2f (single ds_load_b64);
// A^T fragments for GEMM2 preloaded into 128 VGPRs (jb-invariant).
//
// K-paired stripe layout: element (k, n) lives at [(k>>1)*32 + 2n + (k&1)],
// so {row k, row k+1} at column n is an aligned 8-byte pair.
// ---------------------------------------------------------------------------
__global__ void __launch_bounds__(512)
blur_wmma(const float* __restrict__ bel, const float* __restrict__ Ag,
          const float* __restrict__ Atg, float* __restrict__ Gg) {
    __shared__ float m_s[HDIM * WDIM];      // 256 KB: the full map
    __shared__ float t1_s[HDIM * 16];       //  16 KB: T1 stripe (K-paired)
    __shared__ float a_s[HDIM * 16];        //  16 KB: A stripe (K-paired)

    const int map  = blockIdx.x;
    const int tid  = threadIdx.x;
    const float* mg = bel + (size_t)map * HDIM * WDIM;
    float*       gg = Gg  + (size_t)map * HDIM * WDIM;

    const int wv   = tid >> 5;           // wave id 0..15 -> 16-row tile
    const int lane = tid & 31;
    const int n    = lane & 15;          // M (A-frag) / N (B,C,D frags)
    const int kh   = lane >> 4;          // K half-select per ISA 16x4 layout

    // ---- Stage the whole map into LDS (async path when toolchain has it) ----
#if defined(__has_builtin) && __has_builtin(__builtin_amdgcn_global_load_async_to_lds_b128)
    {
        for (int t = 0; t < 32; ++t) {
            const int c = t * 512 + tid;                      // 16-byte chunk id
            glob_i32x4* g = (glob_i32x4*)(uintptr_t)(mg + c * 4);
            lds_i32x4*  l = (lds_i32x4*)(unsigned int)(uintptr_t)(void*)&m_s[c * 4];
            __builtin_amdgcn_global_load_async_to_lds_b128(g, l, 0, 0);
        }
    }
#else
    {
        const float4* src4 = (const float4*)mg;
        float4*       dst4 = (float4*)m_s;
        for (int t = tid; t < (HDIM * WDIM) / 4; t += 512)
            dst4[t] = src4[t];
    }
#endif

    // ---- Preload GEMM2 A^T fragments (jb-invariant) into registers ----
    // a2[kk] holds A^T[16wv+n][4kk+2kh .. +1] == A[4kk+2kh..][16wv+n]
    v2f a2[64];
    {
        const v2f* atrow = (const v2f*)(Atg + (size_t)(16 * wv + n) * WDIM);
#pragma unroll
        for (int kk = 0; kk < 64; ++kk)
            a2[kk] = atrow[2 * kk + kh];
    }

#if defined(__has_builtin) && __has_builtin(__builtin_amdgcn_global_load_async_to_lds_b128)
# if __has_builtin(__builtin_amdgcn_s_wait_asynccnt)
    __builtin_amdgcn_s_wait_asynccnt(0);
# else
    asm volatile("s_wait_asynccnt 0x0" ::: "memory");
# endif
#endif
    __syncthreads();

    const v2f* a_sp  = (const v2f*)a_s;
    const v2f* t1_sp = (const v2f*)t1_s;
    const v2f* m_sp  = (const v2f*)m_s;

    for (int jb = 0; jb < 16; ++jb) {    // 16-column stripes of the output
        // ---- Stage A[:, jb*16 .. +16] into K-paired a_s ----
        {
            const float4* Ag4 = (const float4*)Ag;
            for (int t = tid; t < (HDIM * 16) / 4; t += 512) {
                const int k  = t >> 2;                  // source row of A
                const int c4 = t & 3;                   // float4 within stripe
                const float4 r4 = Ag4[k * 64 + jb * 4 + c4];
                const int base = (k >> 1) * 32 + (k & 1);
                a_s[base + 2 * (4 * c4 + 0)] = r4.x;
                a_s[base + 2 * (4 * c4 + 1)] = r4.y;
                a_s[base + 2 * (4 * c4 + 2)] = r4.z;
                a_s[base + 2 * (4 * c4 + 3)] = r4.w;
            }
        }
        __syncthreads();

        // ---- GEMM1: T1[16wv..+16, stripe] = m(rows) * A(:, stripe) ----
        v8f acc = {0.f,0.f,0.f,0.f,0.f,0.f,0.f,0.f};
        const int mrow = 16 * wv + n;
        for (int k0 = 0; k0 < WDIM; k0 += 4) {
            const int ka = k0 + 2 * kh;                          // even
            const v2f a = m_sp[(mrow * WDIM + ka) >> 1];         // {m[ka], m[ka+1]}
            const v2f b = a_sp[((k0 >> 1) + kh) * 16 + n];       // {A[ka][n], A[ka+1][n]}
            acc = __builtin_amdgcn_wmma_f32_16x16x4_f32(
                false, a, false, b, (short)0, acc, false, false);
        }
#pragma unroll
        for (int v = 0; v < 8; ++v) {
            const int r = 16 * wv + v + 8 * kh;                  // T1 row
            t1_s[(r >> 1) * 32 + 2 * n + (r & 1)] = acc[v];
        }
        __syncthreads();

        // ---- GEMM2: G[16wv..+16, stripe] = A^T(rows) * T1(:, stripe) ----
        v8f acc2 = {0.f,0.f,0.f,0.f,0.f,0.f,0.f,0.f};
#pragma unroll
        for (int kk = 0; kk < 64; ++kk) {
            const v2f b = t1_sp[(2 * kk + kh) * 16 + n];         // ka = 4kk+2kh
            acc2 = __builtin_amdgcn_wmma_f32_16x16x4_f32(
                false, a2[kk], false, b, (short)0, acc2, false, false);
        }
#pragma unroll
        for (int v = 0; v < 8; ++v)
            gg[(16 * wv + v + 8 * kh) * WDIM + jb * 16 + n] = acc2[v];
        __syncthreads();   // t1_s / a_s reused next stripe
    }
}

// ---------------------------------------------------------------------------
// Kernel 3: per-map peak detect on G, argmax of original m over peaks,
// 5x5 centroid around winner, validity logic.  One 256-thread block per map
// (thread t scans row t).
// ---------------------------------------------------------------------------
__global__ void __launch_bounds__(256)
peaks_reduce(const float* __restrict__ bel, const float* __restrict__ Gg,
             float* __restrict__ out) {
    __shared__ int   s_cnt[256];
    __shared__ float s_val[256];
    __shared__ int   s_idx[256];

    const int map = blockIdx.x;
    const int h   = threadIdx.x;
    const float* m = bel + (size_t)map * HDIM * WDIM;
    const float* G = Gg  + (size_t)map * HDIM * WDIM;

    int   cnt   = 0;
    float bestv = -INFINITY;
    int   besti = INT_MAX;

    for (int w = 0; w < WDIM; ++w) {
        const int p = h * WDIM + w;
        const float g = G[p];
        if (g > THRESH) {
            const float up = (h > 0)        ? G[p - WDIM] : 0.0f;
            const float dn = (h < HDIM - 1) ? G[p + WDIM] : 0.0f;
            const float lf = (w > 0)        ? G[p - 1]    : 0.0f;
            const float rt = (w < WDIM - 1) ? G[p + 1]    : 0.0f;
            if (g >= up && g >= dn && g >= lf && g >= rt) {
                ++cnt;
                const float val = m[p];
                if (val > bestv) { bestv = val; besti = p; }  // keep first max
            }
        }
    }
    s_cnt[h] = cnt; s_val[h] = bestv; s_idx[h] = besti;
    __syncthreads();

    for (int s = 128; s > 0; s >>= 1) {
        if (h < s) {
            s_cnt[h] += s_cnt[h + s];
            const float v2 = s_val[h + s];
            const int   i2 = s_idx[h + s];
            if (v2 > s_val[h] || (v2 == s_val[h] && i2 < s_idx[h])) {
                s_val[h] = v2; s_idx[h] = i2;
            }
        }
        __syncthreads();
    }

    if (h == 0) {
        const int npk = s_cnt[0];
        float kx = -999.999f, ky = -999.999f, conf = 0.0f;
        if (npk > 0) {
            const int idx = s_idx[0];
            const int r0 = idx >> 8, c0 = idx & 255;
            const float score = m[idx];
            float S = 0.0f, SY = 0.0f, SX = 0.0f;
            for (int dy = -2; dy <= 2; ++dy) {
                const int r = r0 + dy;
                if (r < 0 || r >= HDIM) continue;
                for (int dx = -2; dx <= 2; ++dx) {
                    const int c = c0 + dx;
                    if (c < 0 || c >= WDIM) continue;
                    const float v = m[r * WDIM + c];
                    S  += v;
                    SY += v * (float)r;
                    SX += v * (float)c;
                }
            }
            const float x = ((S == 0.0f) ? (float)c0 : SX / S) + OFFSET_C;
            const float y = ((S == 0.0f) ? (float)r0 : SY / S) + OFFSET_C;
            const bool valid = (npk == 1) || (score >= SCORE_THRESH);
            if (valid) { kx = x; ky = y; conf = 1.0f; }
        }
        out[map * 3 + 0] = kx;
        out[map * 3 + 1] = ky;
        out[map * 3 + 2] = conf;
    }
}

// ---------------------------------------------------------------------------
// Launch: ws layout = [ A : 256KB | A^T : 256KB | G : 512*65536*4 = 128MB ]
// ---------------------------------------------------------------------------
extern "C" void kernel_launch(void* const* d_in, const int* in_sizes, int n_in,
                              void* d_out, int out_size, void* d_ws, size_t ws_size,
                              hipStream_t stream) {
    (void)in_sizes; (void)n_in; (void)out_size; (void)ws_size;
    const float* bel = (const float*)d_in[0];     // (32,16,256,256) fp32
    float* A  = (float*)d_ws;                     // 256x256 blur matrix
    float* At = A + HDIM * WDIM;                  // its transpose
    float* G  = At + HDIM * WDIM;                 // blurred maps, NMAPS*65536
    float* out = (float*)d_out;                   // (32,16,3) fp32

    build_blur_matrix<<<HDIM, WDIM, 0, stream>>>(A, At);
    blur_wmma<<<NMAPS, 512, 0, stream>>>(bel, A, At, G);
    peaks_reduce<<<NMAPS, 256, 0, stream>>>(bel, G, out);
}